// Unfold_65317862638142
// MI455X (gfx1250) — compile-verified
//
#include <hip/hip_runtime.h>

// Unfold (im2col, k=3, pad=1) on (B=16, C=128, H=64, W=64) fp32.
// Pure data movement: out[(p*9+o)*4096 + h*64 + w] = x[p*4096 + (h+di)*64 + (w+dj)]
// with (di,dj) = (o/3-1, o%3-1), zero outside. One block per plane p.
//
// Traffic = 33.5 MB in + 302 MB out -> HBM-bound (~14.4 us at 23.3 TB/s).
// gfx1250 path: async global->LDS b128 staging (ASYNCcnt) + coalesced b128 stores.

typedef float f32x4 __attribute__((ext_vector_type(4)));
typedef int   v4i   __attribute__((vector_size(16)));  // matches builtin prototype

typedef __attribute__((address_space(1))) v4i* global_v4i_ptr;
typedef __attribute__((address_space(3))) v4i* lds_v4i_ptr;

#define TILE_ELEMS 4096  // 64*64 floats = 16 KB per plane

#if __has_builtin(__builtin_amdgcn_global_load_async_to_lds_b128)
#define USE_ASYNC_LDS 1
#else
#define USE_ASYNC_LDS 0
#endif

__global__ __launch_bounds__(256) void unfold_k3_kernel(const float* __restrict__ x,
                                                        float* __restrict__ out) {
    __shared__ float tile[TILE_ELEMS];

    const int tid = threadIdx.x;
    const int p   = blockIdx.x;  // plane index in [0, B*C)
    const float* src = x + (size_t)p * TILE_ELEMS;

    // ---- Stage the 64x64 plane into LDS (4 x b128 per thread, coalesced) ----
#if USE_ASYNC_LDS
    {
        global_v4i_ptr g = (global_v4i_ptr)src;
        lds_v4i_ptr    l = (lds_v4i_ptr)tile;
#pragma unroll
        for (int it = 0; it < 4; ++it) {
            const int q = it * 256 + tid;  // float4 index of this thread
            __builtin_amdgcn_global_load_async_to_lds_b128(g + q, l + q, 0, 0);
        }
#if __has_builtin(__builtin_amdgcn_s_wait_asynccnt)
        __builtin_amdgcn_s_wait_asynccnt(0);
#else
        asm volatile("s_wait_asynccnt 0" ::: "memory");
#endif
    }
#else
    {
        f32x4* t4 = (f32x4*)tile;
        const f32x4* s4 = (const f32x4*)src;
#pragma unroll
        for (int it = 0; it < 4; ++it) {
            const int q = it * 256 + tid;
            t4[q] = s4[q];
        }
    }
#endif
    __syncthreads();

    // ---- Emit 9 shifted copies, b128 stores, edges zeroed ----
    float* obase = out + (size_t)p * 9 * TILE_ELEMS;
#pragma unroll
    for (int o = 0; o < 9; ++o) {
        const int di = o / 3 - 1;
        const int dj = o % 3 - 1;
        float* op = obase + (size_t)o * TILE_ELEMS;
#pragma unroll
        for (int it = 0; it < 4; ++it) {
            const int q  = it * 256 + tid;   // float4 index in [0,1024)
            const int h  = q >> 4;           // row in [0,64)
            const int w0 = (q & 15) << 2;    // col of first element
            const int h2 = h + di;
            f32x4 v = {0.0f, 0.0f, 0.0f, 0.0f};
            if ((unsigned)h2 < 64u) {
                const float* row = tile + h2 * 64;
#pragma unroll
                for (int j = 0; j < 4; ++j) {
                    const int w2 = w0 + j + dj;
                    v[j] = ((unsigned)w2 < 64u) ? row[w2] : 0.0f;
                }
            }
            *(f32x4*)(op + (q << 2)) = v;   // global_store_b128, coalesced
        }
    }
}

extern "C" void kernel_launch(void* const* d_in, const int* in_sizes, int n_in,
                              void* d_out, int out_size, void* d_ws, size_t ws_size,
                              hipStream_t stream) {
    (void)n_in; (void)out_size; (void)d_ws; (void)ws_size;
    const float* x = (const float*)d_in[0];
    // d_in[1] is the fixed identity kernel -> not needed.
    float* out = (float*)d_out;

    const int planes = in_sizes[0] / TILE_ELEMS;  // B*C = 2048
    dim3 grid(planes), block(256);
    hipLaunchKernelGGL(unfold_k3_kernel, grid, block, 0, stream, x, out);
}